// WeightedDistanceTransform_86474871537856
// MI455X (gfx1250) — compile-verified
//
#include <hip/hip_runtime.h>
#include <math.h>

// Problem geometry (fixed by the reference's setup_inputs).
#define B_ 16
#define C_ 3
#define H_ 256
#define W_ 256
#define NSLICE (B_ * C_)           // 48 independent (b,c) slices
#define SLICE_ELEMS (H_ * W_)      // 65536 floats per slice
#define NTHREADS 1024              // 32 wave32 waves per workgroup
#define LSTRIDE 260                // LDS row stride (floats): rows 16B-aligned, low bank conflict
#define CHUNK 16                   // register accumulators per pass-2 chunk (VGPR-safe)
#define BIGD 512.0f                // H + W, exceeds any achievable 1D distance

#if defined(__gfx1250__) && __has_builtin(__builtin_amdgcn_global_load_async_to_lds_b128)
#define USE_ASYNC_LDS 1
#else
#define USE_ASYNC_LDS 0
#endif

// Builtin parameter types per hipcc diagnostic: v4i pointers (AS(1) source, AS(3) dest).
typedef int v4i __attribute__((ext_vector_type(4)));
typedef __attribute__((address_space(1))) v4i GlobalV4;
typedef __attribute__((address_space(3))) v4i LdsV4;

__global__ __launch_bounds__(NTHREADS, 1)
void wdt_kernel(const float* __restrict__ flood, float* __restrict__ out) {
    // 256 * 260 * 4B = 266,240B tile + 4KB reduction scratch -> fits CDNA5's 320KB WGP LDS.
    __shared__ float tile[H_ * LSTRIDE];
    __shared__ float red[NTHREADS];

    const int tid = threadIdx.x;
    const int slice = blockIdx.x;          // slice = b*C + c
    const int c = slice % C_;
    const float wgt = (c == 0) ? 0.5f : ((c == 1) ? 1.0f : 2.0f);

    const size_t base = (size_t)slice * SLICE_ELEMS;
    const float* __restrict__ src = flood + base;
    float* __restrict__ dst = out + base;

    // ---------------- Stage the 256x256 slice into LDS (padded rows) ----------------
    // Each thread moves 16 aligned float4s; global side fully coalesced.
    #pragma unroll
    for (int k = 0; k < SLICE_ELEMS / (NTHREADS * 4); ++k) {
        const int idx = (k * NTHREADS + tid) * 4;   // float index; multiple of 4, never crosses a row
        const int row = idx >> 8;                   // /W_
        const int col = idx & (W_ - 1);
        const int loff = row * LSTRIDE + col;       // 16B-aligned (LSTRIDE%4==0, col%4==0)
#if USE_ASYNC_LDS
        __builtin_amdgcn_global_load_async_to_lds_b128(
            (GlobalV4*)(src + idx), (LdsV4*)&tile[loff], 0, 0);
#else
        const float4 v = *(const float4*)(src + idx);
        tile[loff + 0] = v.x;
        tile[loff + 1] = v.y;
        tile[loff + 2] = v.z;
        tile[loff + 3] = v.w;
#endif
    }
#if USE_ASYNC_LDS
#if __has_builtin(__builtin_amdgcn_s_wait_asynccnt)
    __builtin_amdgcn_s_wait_asynccnt(0);
#else
    asm volatile("s_wait_asynccnt 0" ::: "memory");
#endif
#endif
    __syncthreads();

    // ---------------- Pass 1: 1D distance-to-zero along each row (in LDS) ----------------
    // Thread t owns row t. Forward value is 0 iff mask==0, so the backward pass can
    // recover the mask from the forward result and write g^2 in place.
    if (tid < H_) {
        const int rb = tid * LSTRIDE;
        float d = BIGD;
        #pragma unroll 4
        for (int w = 0; w < W_; ++w) {
            const float m = tile[rb + w];
            d = (m != 0.0f) ? fminf(d + 1.0f, BIGD) : 0.0f;
            tile[rb + w] = d;                       // forward distance
        }
        d = BIGD;
        #pragma unroll 4
        for (int w = W_ - 1; w >= 0; --w) {
            const float f = tile[rb + w];           // forward dist (0 iff mask==0)
            d = (f != 0.0f) ? fminf(d + 1.0f, BIGD) : 0.0f;
            const float g = fminf(f, d);
            tile[rb + w] = g * g;                   // squared 1D distance
        }
    }
    __syncthreads();

    // ---------------- Pass 2: exact min-plus over rows, per column ----------------
    // Thread -> (column w, 64-row output band q), processed in CHUNK-row register tiles.
    const int w = tid & (W_ - 1);
    const int q = tid >> 8;                         // 0..3
    float lmax = 0.0f;

    for (int ch = 0; ch < 64 / CHUNK; ++ch) {
        const int r0 = q * 64 + ch * CHUNK;
        float acc[CHUNK];
        #pragma unroll
        for (int i = 0; i < CHUNK; ++i) acc[i] = 1.0e30f;

        for (int rp = 0; rp < H_; ++rp) {
            const float g = tile[rp * LSTRIDE + w]; // lanes hit consecutive banks
            const float df0 = (float)(r0 - rp);
            #pragma unroll
            for (int i = 0; i < CHUNK; ++i) {
                const float df = df0 + (float)i;
                acc[i] = fminf(acc[i], fmaf(df, df, g));   // v_add + v_fma + v_min
            }
        }

        #pragma unroll
        for (int i = 0; i < CHUNK; ++i) {
            const float dt = sqrtf(acc[i]) * wgt;
            lmax = fmaxf(lmax, dt);
            dst[(size_t)(r0 + i) * W_ + w] = dt;    // coalesced; re-read below by this thread only
        }
    }

    // ---------------- Per-slice max (block == slice) ----------------
    red[tid] = lmax;
    __syncthreads();
    #pragma unroll
    for (int s = NTHREADS / 2; s > 0; s >>= 1) {
        if (tid < s) red[tid] = fmaxf(red[tid], red[tid + s]);
        __syncthreads();
    }
    const float mx = red[0];
    const float safe = (mx > 0.0f) ? mx : 1.0f;

    // ---------------- Inverted rescale: (mx - dt) / safe ----------------
    // Each thread re-reads exactly the addresses it wrote (same-wave same-address
    // ordering guaranteed). When mx==0, dt==0 everywhere, so result is 0 == dt.
    #pragma unroll 4
    for (int j = 0; j < 64; ++j) {
        const size_t o = (size_t)(q * 64 + j) * W_ + w;
        const float dt = dst[o];
        dst[o] = (mx - dt) / safe;
    }
}

extern "C" void kernel_launch(void* const* d_in, const int* in_sizes, int n_in,
                              void* d_out, int out_size, void* d_ws, size_t ws_size,
                              hipStream_t stream) {
    (void)in_sizes; (void)n_in; (void)d_ws; (void)ws_size; (void)out_size;
    const float* flood = (const float*)d_in[0];
    float* out = (float*)d_out;
    hipLaunchKernelGGL(wdt_kernel, dim3(NSLICE), dim3(NTHREADS), 0, stream, flood, out);
}